// GCNN_33784212750626
// MI455X (gfx1250) — compile-verified
//
#include <hip/hip_runtime.h>
#include <hip/hip_bf16.h>

typedef float v2f __attribute__((ext_vector_type(2)));
typedef float v8f __attribute__((ext_vector_type(8)));

__device__ __forceinline__ v8f wmma_f32x4(v2f a, v2f b, v8f c) {
    // D = A(16x4) * B(4x16) + C(16x16), fp32, wave32
    return __builtin_amdgcn_wmma_f32_16x16x4_f32(
        /*neg_a=*/false, a, /*neg_b=*/false, b,
        /*c_mod=*/(short)0, c, /*reuse_a=*/false, /*reuse_b=*/false);
}

// ---------------- degree / norm kernels ----------------

__global__ void init_norm_kernel(float* __restrict__ ns, float* __restrict__ nd, int N) {
    int i = blockIdx.x * blockDim.x + threadIdx.x;
    if (i < N) { ns[i] = 1.0f; nd[i] = 1.0f; }  // self-loop contributes 1 to both degrees
}

__global__ void deg_edges_kernel(const int* __restrict__ src, const int* __restrict__ dst,
                                 float* __restrict__ ns, float* __restrict__ nd, int E) {
    int e = blockIdx.x * blockDim.x + threadIdx.x;
    if (e < E) {
        atomicAdd(&ns[src[e]], 1.0f);
        atomicAdd(&nd[dst[e]], 1.0f);
    }
}

__global__ void finalize_norm_kernel(float* __restrict__ ns, float* __restrict__ nd, int N) {
    int i = blockIdx.x * blockDim.x + threadIdx.x;
    if (i < N) {
        ns[i] = rsqrtf(ns[i]);   // deg >= 1 always (self loop), clip is a no-op
        nd[i] = rsqrtf(nd[i]);
    }
}

// bufA = features * norm_src ; bufB = same (self-loop init of aggregate)
__global__ void scale_init_kernel(const float* __restrict__ feat, const float* __restrict__ ns,
                                  float* __restrict__ bufA, float* __restrict__ bufB, int total) {
    int t = blockIdx.x * blockDim.x + threadIdx.x;
    if (t < total) {
        float v = feat[t] * ns[t >> 6];
        bufA[t] = v;
        bufB[t] = v;
    }
}

// ---------------- edge scatter-add (64 floats per edge, 16 lanes/edge, float4 each) ----

__global__ void edge_agg_kernel(const int* __restrict__ src, const int* __restrict__ dst,
                                const float* __restrict__ hs, float* __restrict__ agg, int E) {
    int gid = blockIdx.x * blockDim.x + threadIdx.x;
    int e = gid >> 4;
    if (e >= E) return;
    int part = (gid & 15) << 2;                    // 0,4,...,60
    int s = src[e];
    int d = dst[e];
    const float4 v = *reinterpret_cast<const float4*>(hs + s * 64 + part);
    float* p = agg + d * 64 + part;
    atomicAdd(p + 0, v.x);
    atomicAdd(p + 1, v.y);
    atomicAdd(p + 2, v.z);
    atomicAdd(p + 3, v.w);
}

// ---------------- layer 1 GEMM: (X .* nd) @ W1 + b1, relu, .* ns  -> outA and outB ----
// Each wave owns rows [wave*16, wave*16+16) and all 64 output columns (4 WMMA tiles).

__global__ void gemm1_kernel(const float* __restrict__ X, const float* __restrict__ nd,
                             const float* __restrict__ ns, const float* __restrict__ W,
                             const float* __restrict__ bias,
                             float* __restrict__ outA, float* __restrict__ outB, int N) {
    int wave = blockIdx.x * (blockDim.x >> 5) + (threadIdx.x >> 5);
    int row0 = wave << 4;
    if (row0 >= N) return;                         // wave-uniform: EXEC stays all-ones
    int lane = threadIdx.x & 31;
    int half = lane >> 4;                          // 0: lanes 0-15, 1: lanes 16-31
    int lm   = lane & 15;

    v8f c0 = {}, c1 = {}, c2 = {}, c3 = {};

    int arow = row0 + lm;
    if (arow >= N) arow = N - 1;                   // clamp keeps EXEC full (no-op: N%16==0)
    float ndv = nd[arow];
    const float* Ap = X + arow * 64;

#pragma unroll
    for (int k0 = 0; k0 < 64; k0 += 4) {
        int kA = k0 + half * 2;                    // lanes 0-15: K=k0,k0+1; 16-31: K=k0+2,k0+3
        v2f a;
        a.x = Ap[kA] * ndv;
        a.y = Ap[kA + 1] * ndv;
        const float* Wr0 = W + kA * 64;
        const float* Wr1 = W + (kA + 1) * 64;
        v2f b;
        b.x = Wr0[lm];      b.y = Wr1[lm];      c0 = wmma_f32x4(a, b, c0);
        b.x = Wr0[16 + lm]; b.y = Wr1[16 + lm]; c1 = wmma_f32x4(a, b, c1);
        b.x = Wr0[32 + lm]; b.y = Wr1[32 + lm]; c2 = wmma_f32x4(a, b, c2);
        b.x = Wr0[48 + lm]; b.y = Wr1[48 + lm]; c3 = wmma_f32x4(a, b, c3);
    }

#pragma unroll
    for (int r = 0; r < 8; r++) {
        int row = row0 + r + half * 8;             // C/D layout: VGPR r = rows r / r+8
        if (row >= N) continue;                    // after all WMMAs; divergence is fine
        float nsr = ns[row];
        float* pa = outA + row * 64;
        float* pb = outB + row * 64;
        float v;
        v = c0[r] + bias[lm];      v = v > 0.f ? v : 0.f; v *= nsr; pa[lm]      = v; pb[lm]      = v;
        v = c1[r] + bias[16 + lm]; v = v > 0.f ? v : 0.f; v *= nsr; pa[16 + lm] = v; pb[16 + lm] = v;
        v = c2[r] + bias[32 + lm]; v = v > 0.f ? v : 0.f; v *= nsr; pa[32 + lm] = v; pb[32 + lm] = v;
        v = c3[r] + bias[48 + lm]; v = v > 0.f ? v : 0.f; v *= nsr; pa[48 + lm] = v; pb[48 + lm] = v;
    }
}

// ---------------- layer 2 GEMM: (X .* nd) @ W2 + b2 -> out (N x 32) ----------------

__global__ void gemm2_kernel(const float* __restrict__ X, const float* __restrict__ nd,
                             const float* __restrict__ W, const float* __restrict__ bias,
                             float* __restrict__ out, int N) {
    int wave = blockIdx.x * (blockDim.x >> 5) + (threadIdx.x >> 5);
    int row0 = wave << 4;
    if (row0 >= N) return;
    int lane = threadIdx.x & 31;
    int half = lane >> 4;
    int lm   = lane & 15;

    v8f c0 = {}, c1 = {};

    int arow = row0 + lm;
    if (arow >= N) arow = N - 1;
    float ndv = nd[arow];
    const float* Ap = X + arow * 64;

#pragma unroll
    for (int k0 = 0; k0 < 64; k0 += 4) {
        int kA = k0 + half * 2;
        v2f a;
        a.x = Ap[kA] * ndv;
        a.y = Ap[kA + 1] * ndv;
        const float* Wr0 = W + kA * 32;
        const float* Wr1 = W + (kA + 1) * 32;
        v2f b;
        b.x = Wr0[lm];      b.y = Wr1[lm];      c0 = wmma_f32x4(a, b, c0);
        b.x = Wr0[16 + lm]; b.y = Wr1[16 + lm]; c1 = wmma_f32x4(a, b, c1);
    }

#pragma unroll
    for (int r = 0; r < 8; r++) {
        int row = row0 + r + half * 8;
        if (row >= N) continue;
        float* po = out + row * 32;
        po[lm]      = c0[r] + bias[lm];
        po[16 + lm] = c1[r] + bias[16 + lm];
    }
}

// ---------------- launch ----------------

extern "C" void kernel_launch(void* const* d_in, const int* in_sizes, int n_in,
                              void* d_out, int out_size, void* d_ws, size_t ws_size,
                              hipStream_t stream) {
    const float* feat = (const float*)d_in[0];
    const int*   src  = (const int*)d_in[1];
    const int*   dst  = (const int*)d_in[2];
    const float* W1   = (const float*)d_in[3];
    const float* b1   = (const float*)d_in[4];
    const float* W2   = (const float*)d_in[5];
    const float* b2   = (const float*)d_in[6];
    float* out = (float*)d_out;

    const int N = in_sizes[0] / 64;   // 100000
    const int E = in_sizes[1];        // 1600000

    // workspace layout: ns[N] | nd[N] | bufA[N*64] | bufB[N*64]  (~52 MB)
    float* ns   = (float*)d_ws;
    float* nd   = ns + N;
    float* bufA = nd + N;
    float* bufB = bufA + (size_t)N * 64;

    const int TPB = 256;

    // degrees (as float, starting at 1.0 for the self loop) -> rsqrt norms
    init_norm_kernel<<<(N + TPB - 1) / TPB, TPB, 0, stream>>>(ns, nd, N);
    deg_edges_kernel<<<(E + TPB - 1) / TPB, TPB, 0, stream>>>(src, dst, ns, nd, E);
    finalize_norm_kernel<<<(N + TPB - 1) / TPB, TPB, 0, stream>>>(ns, nd, N);

    // layer 1 input: bufA = feat .* ns ; bufB = self-loop init of aggregate
    int total = N * 64;
    scale_init_kernel<<<(total + TPB - 1) / TPB, TPB, 0, stream>>>(feat, ns, bufA, bufB, total);

    // scatter-add over real edges
    int ethreads = E * 16;
    edge_agg_kernel<<<(ethreads + TPB - 1) / TPB, TPB, 0, stream>>>(src, dst, bufA, bufB, E);

    // GEMM1 (WMMA), fused bias+relu+norm_src; writes next gather source (bufA) and
    // next aggregate's self-loop init (bufB, overwriting its own consumed rows)
    int waves = (N + 15) / 16;                       // 6250
    int wavesPerBlock = TPB / 32;                    // 8
    int gblocks = (waves + wavesPerBlock - 1) / wavesPerBlock;
    gemm1_kernel<<<gblocks, TPB, 0, stream>>>(bufB, nd, ns, W1, b1, bufA, bufB, N);

    // scatter-add for layer 2
    edge_agg_kernel<<<(ethreads + TPB - 1) / TPB, TPB, 0, stream>>>(src, dst, bufA, bufB, E);

    // GEMM2 (WMMA), fused bias -> final output [N, 32]
    gemm2_kernel<<<gblocks, TPB, 0, stream>>>(bufB, nd, W2, b2, out, N);
}